// Router_34711925686735
// MI455X (gfx1250) — compile-verified
//
#include <hip/hip_runtime.h>
#include <hip/hip_bf16.h>
#include <math.h>

// MoE router: logits = x @ W^T * 1/sqrt(D); top-8 by (logits+bias); softmax over
// unbiased logits restricted to the top-8.
// Memory-bound: 128 MB of x @ 23.3 TB/s ~= 5.5us floor. Full fp32 precision via
// V_WMMA_F32_16X16X4_F32. K split across wave pairs (2048 waves total for
// latency hiding on a ~1000-SIMD part); top-k scan parallelized over both
// lane halves of the wave32.

#define D_MODEL 2048
#define N_EXPERTS 64
#define TOP_K 8
#define T_TOTAL 16384
#define K_SPLIT 2
#define GROUPS_PER_BLOCK 4                       // token groups per block
#define WAVES_PER_BLOCK (GROUPS_PER_BLOCK * K_SPLIT)
#define TOKENS_PER_WAVE 16
#define K_HALF (D_MODEL / K_SPLIT)               // 1024
#define ROW_PITCH 68   // LDS row pitch in floats: (4*l + e) % 64 conflict-free

typedef __attribute__((ext_vector_type(2))) float v2f;
typedef __attribute__((ext_vector_type(8))) float v8f;

__global__ __launch_bounds__(WAVES_PER_BLOCK * 32, 1) void router_wmma_kernel(
    const float* __restrict__ x,
    const float* __restrict__ W,
    const float* __restrict__ rbias,
    float* __restrict__ out_w,
    int*   __restrict__ out_i)
{
    __shared__ float s_bias[N_EXPERTS];
    __shared__ float s_tile[GROUPS_PER_BLOCK][TOKENS_PER_WAVE * ROW_PITCH];

    const int tid  = threadIdx.x;
    const int lane = tid & 31;
    const int wv   = tid >> 5;
    const int g    = wv & (GROUPS_PER_BLOCK - 1);  // token group within block
    const int kh   = wv / GROUPS_PER_BLOCK;        // K half: 0 or 1

    if (tid < N_EXPERTS) s_bias[tid] = rbias[tid];
    __syncthreads();

    const int tokenBase = (blockIdx.x * GROUPS_PER_BLOCK + g) * TOKENS_PER_WAVE;
    const int lh   = lane >> 4;   // 0: lanes 0-15 (K=0,1), 1: lanes 16-31 (K=2,3)
    const int lrow = lane & 15;

    // A: 16x4 fp32 tile of x rows; B: 4x16 fp32 tile of W rows (same addressing,
    // since W is [out,in] row-major and B column n == W row n).
    const size_t kOff = (size_t)kh * K_HALF + 2 * lh;
    const float* aptr  = x + (size_t)(tokenBase + lrow) * D_MODEL + kOff;
    const float* bptr0 = W + (size_t)(lrow) * D_MODEL + kOff;
    const float* bptr1 = bptr0 + (size_t)16 * D_MODEL;
    const float* bptr2 = bptr0 + (size_t)32 * D_MODEL;
    const float* bptr3 = bptr0 + (size_t)48 * D_MODEL;

    v8f acc0 = {}, acc1 = {}, acc2 = {}, acc3 = {};

    #pragma unroll 4
    for (int k = 0; k < K_HALF; k += 4) {
        v2f a  = *(const v2f*)(aptr  + k);
        v2f b0 = *(const v2f*)(bptr0 + k);
        v2f b1 = *(const v2f*)(bptr1 + k);
        v2f b2 = *(const v2f*)(bptr2 + k);
        v2f b3 = *(const v2f*)(bptr3 + k);
        // 8 args: (neg_a, A, neg_b, B, c_mod, C, reuse_a, reuse_b)
        acc0 = __builtin_amdgcn_wmma_f32_16x16x4_f32(false, a, false, b0, (short)0, acc0, false, false);
        acc1 = __builtin_amdgcn_wmma_f32_16x16x4_f32(false, a, false, b1, (short)0, acc1, false, false);
        acc2 = __builtin_amdgcn_wmma_f32_16x16x4_f32(false, a, false, b2, (short)0, acc2, false, false);
        acc3 = __builtin_amdgcn_wmma_f32_16x16x4_f32(false, a, false, b3, (short)0, acc3, false, false);
    }

    // C/D layout: VGPR r = row (lh*8 + r), lane%16 = col within 16-expert tile.
    float* row = s_tile[g];

    if (kh == 1) {
        // Store partial sums for the upper K half.
        #pragma unroll
        for (int r = 0; r < 8; ++r) {
            const int m = lh * 8 + r;
            row[m * ROW_PITCH + ( 0 + lrow)] = acc0[r];
            row[m * ROW_PITCH + (16 + lrow)] = acc1[r];
            row[m * ROW_PITCH + (32 + lrow)] = acc2[r];
            row[m * ROW_PITCH + (48 + lrow)] = acc3[r];
        }
    }
    __syncthreads();
    if (kh == 1) return;   // kh==0 waves finish the job (wave-local from here on)

    // Combine partials and write final unscaled, unbiased logits tile back to LDS.
    #pragma unroll
    for (int r = 0; r < 8; ++r) {
        const int m = lh * 8 + r;
        row[m * ROW_PITCH + ( 0 + lrow)] += acc0[r];
        row[m * ROW_PITCH + (16 + lrow)] += acc1[r];
        row[m * ROW_PITCH + (32 + lrow)] += acc2[r];
        row[m * ROW_PITCH + (48 + lrow)] += acc3[r];
    }
    // Same-wave LDS RAW below: compiler inserts s_wait_dscnt.

    // ---- Top-8 + masked softmax, parallel over both lane halves. ----
    // Lane pair (lrow, lrow+16) owns token lrow; half lh scans experts
    // [32*lh, 32*lh+32). Per-pass argmax merged across halves via shfl_xor(16).
    const float scale = 0.0220970869120796101f; // 1/sqrt(2048)
    const int token   = tokenBase + lrow;
    const int eBase   = 32 * lh;

    float v[32];
    #pragma unroll
    for (int j = 0; j < 32; ++j)
        v[j] = row[lrow * ROW_PITCH + eBase + j] * scale;

    float topv[TOP_K];
    int   topi[TOP_K];
    unsigned int used = 0u;  // local 32-expert mask for this half
    #pragma unroll
    for (int k = 0; k < TOP_K; ++k) {
        float best = -INFINITY, bestu = 0.0f;
        int bi = eBase;
        #pragma unroll
        for (int j = 0; j < 32; ++j) {
            const int e = eBase + j;
            const float b = v[j] + s_bias[e];
            const bool ok = (b > best) && (((used >> j) & 1u) == 0u);
            best  = ok ? b    : best;
            bestu = ok ? v[j] : bestu;
            bi    = ok ? e    : bi;
        }
        // Merge argmax across the two halves; ties -> lower expert index
        // (matches jax.lax.top_k tie-breaking).
        const float obest  = __shfl_xor(best,  16);
        const float obestu = __shfl_xor(bestu, 16);
        const int   obi    = __shfl_xor(bi,    16);
        const bool takeOther = (obest > best) || ((obest == best) && (obi < bi));
        best  = takeOther ? obest  : best;
        bestu = takeOther ? obestu : bestu;
        bi    = takeOther ? obi    : bi;
        if ((bi >> 5) == lh) used |= 1u << (bi & 31);
        topv[k] = bestu;   // UNBIASED scaled logit for softmax
        topi[k] = bi;
    }

    // Softmax over the 8 selected unbiased logits (== masked softmax).
    float m = topv[0];
    #pragma unroll
    for (int k = 1; k < TOP_K; ++k) m = fmaxf(m, topv[k]);
    float ex[TOP_K], s = 0.0f;
    #pragma unroll
    for (int k = 0; k < TOP_K; ++k) { ex[k] = __expf(topv[k] - m); s += ex[k]; }
    const float inv = 1.0f / s;

    if (lh == 0) {
        #pragma unroll
        for (int k = 0; k < TOP_K; ++k) {
            out_w[token * TOP_K + k] = ex[k] * inv;
            out_i[token * TOP_K + k] = topi[k];
        }
    }
}

extern "C" void kernel_launch(void* const* d_in, const int* in_sizes, int n_in,
                              void* d_out, int out_size, void* d_ws, size_t ws_size,
                              hipStream_t stream) {
    (void)in_sizes; (void)n_in; (void)out_size; (void)d_ws; (void)ws_size;
    const float* x  = (const float*)d_in[0];
    const float* W  = (const float*)d_in[1];
    const float* rb = (const float*)d_in[2];
    float* out_w = (float*)d_out;
    int*   out_i = (int*)((float*)d_out + (size_t)T_TOTAL * TOP_K);

    dim3 grid(T_TOTAL / (GROUPS_PER_BLOCK * TOKENS_PER_WAVE)); // 256 blocks
    router_wmma_kernel<<<grid, WAVES_PER_BLOCK * 32, 0, stream>>>(x, W, rb, out_w, out_i);
}